// QuantumPhysicsEngine_53188874994234
// MI455X (gfx1250) — compile-verified
//
#include <hip/hip_runtime.h>
#include <math.h>

// ---------------------------------------------------------------------------
// QuantumPhysicsEngine: normalization sum over 16M grid points (bandwidth
// bound, ~128 MB -> ~5.5us floor @ 23.3 TB/s) + O(100) sequential Langevin
// walk. Wave-level reduction done with V_WMMA_F32_16X16X4_F32 (B = ones)
// for a deterministic f32 cross-lane sum.
// ---------------------------------------------------------------------------

#define GRID_N      16777216
#define X_MIN_F     (-10.0f)
#define DX_F        ((float)(20.0 / (double)(GRID_N - 1)))
#define DT_F        0.01f
#define STEPS_N     100
#define SQRT_2DT_F  0.14142135623730951f

#define RED_BLOCKS  4096
#define RED_THREADS 256
#define CHUNKS_PER_THREAD 8   // float4 chunks; 4096*256*8*4 floats = 33,554,432

typedef __attribute__((ext_vector_type(2))) float v2f;
typedef __attribute__((ext_vector_type(8))) float v8f;

__device__ __forceinline__ float nan_fix(float v) {
    // jnp.nan_to_num(psi, nan=0.0, posinf=1.0, neginf=-1.0)
    if (v != v) return 0.0f;
    if (v == INFINITY) return 1.0f;
    if (v == -INFINITY) return -1.0f;
    return v;
}

// Deterministic wave32 sum of each lane's accumulator via WMMA:
// A[m][0] = acc(lane m), A[m][2] = acc(lane 16+m), other K = 0, B = ones.
// => D[m][n] = acc[m] + acc[m+16] for every n.
// Sum of the 8 D VGPRs in a lane = rows 0..7 (lanes 0-15) or 8..15 (lanes
// 16-31); one shfl_xor(16) add produces the full wave sum in every lane.
__device__ __forceinline__ float wave_sum_wmma(float acc) {
    v2f a; a[0] = acc;  a[1] = 0.0f;
    v2f b; b[0] = 1.0f; b[1] = 1.0f;
    v8f c = {};
    v8f d = __builtin_amdgcn_wmma_f32_16x16x4_f32(
        /*neg_a=*/false, a, /*neg_b=*/false, b,
        /*c_mod=*/(short)0, c, /*reuse_a=*/false, /*reuse_b=*/false);
    float r = ((d[0] + d[1]) + (d[2] + d[3])) + ((d[4] + d[5]) + (d[6] + d[7]));
    r += __shfl_xor(r, 16, 32);
    return r;
}

// ---------------------------------------------------------------------------
// Pass 1: streaming reduction of prob = psi0^2 + psi1^2 (nan_to_num'd).
// One partial sum per block, deterministic order throughout.
// ---------------------------------------------------------------------------
__global__ __launch_bounds__(RED_THREADS)
void qpe_reduce_kernel(const float4* __restrict__ psi4, float* __restrict__ partials) {
    __shared__ float smem[RED_THREADS / 32];

    float acc = 0.0f;
    const size_t base = (size_t)blockIdx.x * (RED_THREADS * CHUNKS_PER_THREAD);
#pragma unroll
    for (int i = 0; i < CHUNKS_PER_THREAD; ++i) {
        float4 v = psi4[base + (size_t)i * RED_THREADS + threadIdx.x];
        float a0 = nan_fix(v.x), a1 = nan_fix(v.y);
        float b0 = nan_fix(v.z), b1 = nan_fix(v.w);
        acc += fmaf(a0, a0, a1 * a1);
        acc += fmaf(b0, b0, b1 * b1);
    }

    // wave32 reduction via v_wmma_f32_16x16x4_f32 (deterministic)
    float wsum = wave_sum_wmma(acc);

    const int lane = threadIdx.x & 31;
    const int wave = threadIdx.x >> 5;
    if (lane == 0) smem[wave] = wsum;
    __syncthreads();

    if (threadIdx.x == 0) {
        float t = 0.0f;
#pragma unroll
        for (int w = 0; w < RED_THREADS / 32; ++w) t += smem[w];
        partials[blockIdx.x] = t;
    }
}

// ---------------------------------------------------------------------------
// Pass 2: deterministic reduction of 4096 partials, then the sequential
// 100-step Langevin walk on thread 0. Output: x[c_final].
// ---------------------------------------------------------------------------
__device__ __forceinline__ float prob_clipped(const float* __restrict__ psi,
                                              int idx, float invZ) {
    float p0 = nan_fix(psi[2 * (size_t)idx]);
    float p1 = nan_fix(psi[2 * (size_t)idx + 1]);
    float raw = fmaf(p0, p0, p1 * p1);
    float pn = raw * invZ;
    return pn > 1e-12f ? pn : 1e-12f;
}

__global__ __launch_bounds__(256)
void qpe_walk_kernel(const float* __restrict__ psi,
                     const float* __restrict__ noise,
                     const float* __restrict__ x,
                     const int* __restrict__ c0,
                     const float* __restrict__ partials,
                     float* __restrict__ out) {
    __shared__ float smem[256];

    // fixed-order accumulation: thread t sums partials[t + 256k]
    float t = 0.0f;
#pragma unroll
    for (int k = 0; k < RED_BLOCKS / 256; ++k)
        t += partials[threadIdx.x + k * 256];
    smem[threadIdx.x] = t;
    __syncthreads();

    // fixed-order tree reduce
    for (int s = 128; s > 0; s >>= 1) {
        if (threadIdx.x < s) smem[threadIdx.x] += smem[threadIdx.x + s];
        __syncthreads();
    }

    if (threadIdx.x == 0) {
        const float Z = smem[0] * DX_F + 1e-8f;
        const float invZ = 1.0f / Z;
        const float inv2dx = 1.0f / (2.0f * DX_F);
        const float invdx = 1.0f / DX_F;

        int c = c0[0];
        for (int step = 0; step < STEPS_N; ++step) {
            float s = 0.0f;
            if (c >= 1 && c <= GRID_N - 2) {
                float pA = prob_clipped(psi, c + 1, invZ);
                float pB = prob_clipped(psi, c - 1, invZ);
                s = (logf(pA) - logf(pB)) * inv2dx;
                if (!isfinite(s)) s = 0.0f;   // nan_to_num(score)
            }
            float move = fmaf(s, DT_F, noise[step] * SQRT_2DT_F);
            if (!isfinite(move)) move = 0.0f;
            float ratio = move * invdx;
            // saturate to grid span: any |shift| >= GRID clips to an endpoint
            // anyway, and this avoids UB in the float->int conversion.
            if (ratio > (float)GRID_N)  ratio = (float)GRID_N;
            if (ratio < -(float)GRID_N) ratio = -(float)GRID_N;
            int shift = (int)ratio;            // truncation toward zero
            long long cn = (long long)c + (long long)shift;
            if (cn < 0) cn = 0;
            if (cn > GRID_N - 1) cn = GRID_N - 1;
            c = (int)cn;
        }
        out[0] = x[c];
    }
}

// ---------------------------------------------------------------------------
extern "C" void kernel_launch(void* const* d_in, const int* in_sizes, int n_in,
                              void* d_out, int out_size, void* d_ws, size_t ws_size,
                              hipStream_t stream) {
    const float* psi   = (const float*)d_in[0];   // [GRID,2] f32
    const float* noise = (const float*)d_in[1];   // [100]    f32
    const float* x     = (const float*)d_in[2];   // [GRID,1] f32
    const int*   c0    = (const int*)d_in[3];     // scalar   i32
    float* out = (float*)d_out;
    float* partials = (float*)d_ws;               // 4096 floats

    qpe_reduce_kernel<<<RED_BLOCKS, RED_THREADS, 0, stream>>>(
        (const float4*)psi, partials);
    qpe_walk_kernel<<<1, 256, 0, stream>>>(psi, noise, x, c0, partials, out);
}